// GCNSingleLayer_13280038879717
// MI455X (gfx1250) — compile-verified
//
#include <hip/hip_runtime.h>

typedef __attribute__((ext_vector_type(2))) float v2f;
typedef __attribute__((ext_vector_type(4))) float v4f;
typedef __attribute__((ext_vector_type(8))) float v8f;

#define N_NODES 50000
#define N_EDGES 800000
#define D       128   // D_IN == D_HID == 128

// ---------------------------------------------------------------------------
// Kernel 1: out[n, :] = b[:]   (bias broadcast; output starts as bias,
// scatter-add accumulates messages on top). float4 streaming stores.
// total float4 elements = N_NODES * D / 4 = 1,600,000 -> 6250 blocks of 256.
// ---------------------------------------------------------------------------
__global__ void gcn_init_bias(const float* __restrict__ b,
                              float* __restrict__ out) {
  int i = blockIdx.x * blockDim.x + threadIdx.x;        // float4 index
  const v4f* b4 = (const v4f*)b;
  ((v4f*)out)[i] = b4[i & (D / 4 - 1)];                 // i % 32 -> column group
}

// ---------------------------------------------------------------------------
// Kernel 2: H = X @ W^T via V_WMMA_F32_16X16X4_F32 (fp32 matrix cores).
//   H[m][n] = sum_k X[m][k] * W[n][k]   (so B[k][n] = W[n][k])
// Block = 256 threads = 8 waves. Block b covers rows [16b, 16b+16).
// Wave w covers cols [16w, 16w+16). K=128 consumed in 32 steps of 4.
//
// A 16x4 f32 layout (ISA 7.12.2): lanes 0-15 hold M=lane, VGPR0=K, VGPR1=K+1;
// lanes 16-31 hold the K+2/K+3 halves. B 4x16 mirrors (rows=K striped across
// lane halves, N in lanes). C/D: VGPR r -> (M=r, N=lane) / (M=r+8, N=lane-16).
// ---------------------------------------------------------------------------
__global__ void gcn_gemm_wmma(const float* __restrict__ X,
                              const float* __restrict__ W,
                              float* __restrict__ H) {
  const int lane  = threadIdx.x & 31;
  const int wave  = threadIdx.x >> 5;        // 0..7 -> N tile
  const int mbase = blockIdx.x * 16;         // 3125 blocks cover 50000 rows
  const int nbase = wave * 16;
  const int half  = lane >> 4;               // 0: K,K+1   1: K+2,K+3
  const int l     = lane & 15;

  const float* __restrict__ xrow = X + (size_t)(mbase + l) * D; // A row M=l
  const float* __restrict__ wrow = W + (size_t)(nbase + l) * D; // B col N=l

  v8f acc = {};
#pragma unroll
  for (int k = 0; k < D; k += 4) {
    v2f a = *(const v2f*)(xrow + k + 2 * half);   // X[m][k+2h], X[m][k+2h+1]
    v2f bm = *(const v2f*)(wrow + k + 2 * half);  // W[n][k+2h] = B[k+2h][n]
    acc = __builtin_amdgcn_wmma_f32_16x16x4_f32(
        /*neg_a=*/false, a, /*neg_b=*/false, bm,
        /*c_mod=*/(short)0, acc, /*reuse_a=*/false, /*reuse_b=*/false);
  }

  // Store D-tile: VGPR r -> row mbase + r + 8*half, col nbase + l
  float* __restrict__ hrow = H + (size_t)(mbase + 8 * half) * D + nbase + l;
#pragma unroll
  for (int r = 0; r < 8; ++r) {
    hrow[(size_t)r * D] = acc[r];
  }
}

// ---------------------------------------------------------------------------
// Kernel 3: scatter-add.  out[dst[e], :] += H[src[e], :]
// One 32-lane wave per edge; each lane moves a float4 (128 floats/row).
// 8 edges per 256-thread block -> 100,000 blocks.
// Atomic f32 adds resolve in L2 (out is 25.6 MB << 192 MB L2).
// ---------------------------------------------------------------------------
__global__ void gcn_scatter(const int* __restrict__ ei,
                            const float* __restrict__ H,
                            float* __restrict__ out) {
  const int e    = blockIdx.x * 8 + (threadIdx.x >> 5);
  const int lane = threadIdx.x & 31;
  const int src  = ei[e];
  const int dst  = ei[N_EDGES + e];

  v4f m = ((const v4f*)(H + (size_t)src * D))[lane];
  float* o = out + (size_t)dst * D + lane * 4;
  atomicAdd(o + 0, m.x);
  atomicAdd(o + 1, m.y);
  atomicAdd(o + 2, m.z);
  atomicAdd(o + 3, m.w);
}

// ---------------------------------------------------------------------------
// Launch: inputs are {x, edge_index, W, b}; output [N_NODES, D] fp32.
// d_ws holds H (50000*128*4 = 25.6 MB).
// ---------------------------------------------------------------------------
extern "C" void kernel_launch(void* const* d_in, const int* in_sizes, int n_in,
                              void* d_out, int out_size, void* d_ws, size_t ws_size,
                              hipStream_t stream) {
  const float* x  = (const float*)d_in[0];
  const int*   ei = (const int*)d_in[1];
  const float* W  = (const float*)d_in[2];
  const float* b  = (const float*)d_in[3];
  float* out = (float*)d_out;
  float* H   = (float*)d_ws;

  // 1) out = bias broadcast
  gcn_init_bias<<<(N_NODES * D / 4) / 256, 256, 0, stream>>>(b, out);

  // 2) H = x @ W^T  (WMMA fp32)
  gcn_gemm_wmma<<<N_NODES / 16, 256, 0, stream>>>(x, W, H);

  // 3) out[dst] += H[src]
  gcn_scatter<<<N_EDGES / 8, 256, 0, stream>>>(ei, H, out);
}